// LightweightDynamicLMSAWithAsymmetricDilatedConv_71347996721818
// MI455X (gfx1250) — compile-verified
//
#include <hip/hip_runtime.h>
#include <math.h>

#define CG 64
#define HH 80
#define WW 80
#define NGRP 64          // B * FACTOR = 8 * 8
#define RROWS 8          // output rows per block
#define HALO 6           // max vertical halo: DIL * (7/2) = 6
#define LDSROWS (RROWS + 2 * HALO)   // 20
#define STR 66           // padded channel stride (halves): 132B = 33 banks, odd -> conflict-free
#define BN_EPS 1e-5f

typedef __attribute__((ext_vector_type(2))) float v2f;
typedef __attribute__((ext_vector_type(8))) float v8f;
typedef __attribute__((ext_vector_type(2))) _Float16 v2h;

struct Params24 { const float* p[24]; };  // 3 branches x {dw1,bn1,dw2,bn2,pwr,bnr,pwe,bne}

// ---------------------------------------------------------------- zero scratch
__global__ void zero_kernel(float* __restrict__ p, int n) {
  int i = blockIdx.x * blockDim.x + threadIdx.x;
  if (i < n) p[i] = 0.f;
}

// ------------------------------------------------- expert selection per group
__global__ void select_kernel(const float* __restrict__ x,
                              const float* __restrict__ sel_w,
                              const float* __restrict__ sel_b,
                              int* __restrict__ idx) {
  __shared__ float part[256];
  __shared__ float pooled[CG];
  int n = blockIdx.x, t = threadIdx.x;
  int c = t & 63, q = t >> 6;
  const float* base = x + ((size_t)(n * CG + c)) * (HH * WW) + q * 1600;
  float s = 0.f;
  for (int i = 0; i < 1600; ++i) s += base[i];
  part[t] = s;
  __syncthreads();
  if (t < CG)
    pooled[t] = (part[t] + part[t + 64] + part[t + 128] + part[t + 192]) * (1.f / 6400.f);
  __syncthreads();
  if (t == 0) {
    float lg[3];
    for (int kk = 0; kk < 3; ++kk) {
      float a = sel_b[kk];
      for (int cc = 0; cc < CG; ++cc) a += pooled[cc] * sel_w[kk * CG + cc];
      lg[kk] = a;
    }
    float m = fmaxf(lg[0], fmaxf(lg[1], lg[2]));
    float e0 = __expf(lg[0] - m), e1 = __expf(lg[1] - m), e2 = __expf(lg[2] - m);
    float inv = 1.f / (e0 + e1 + e2);
    // deterministic uniform from (key=42, n) -- splitmix64
    unsigned long long z = 42ULL * 0x9E3779B97F4A7C15ULL + (unsigned long long)n + 1ULL;
    z ^= z >> 30; z *= 0xBF58476D1CE4E5B9ULL;
    z ^= z >> 27; z *= 0x94D049BB133111EBULL;
    z ^= z >> 31;
    float u = (float)(z >> 40) * (1.f / 16777216.f);
    float c0 = e0 * inv, c1 = c0 + e1 * inv;
    idx[n] = (u < c0) ? 0 : ((u < c1) ? 1 : 2);
  }
}

// ---------------------------------------- fused selected-expert branch kernel
// grid: (10 row tiles, 64 groups), block: 256 threads (8 waves, wave32)
__global__ void __launch_bounds__(256)
branch_kernel(const float* __restrict__ x, Params24 P, const int* __restrict__ idx,
              float* __restrict__ out, float* __restrict__ chansum_g) {
  extern __shared__ char smem[];
  // h1: [LDSROWS][WW][STR] f16, channel-interleaved (c fastest) so a lane's
  // (c0,c1) pair is one aligned b32 LDS load; STR=66 kills bank conflicts.
  _Float16* h1 = (_Float16*)smem;
  float* fw = (float*)(smem + (size_t)LDSROWS * WW * STR * sizeof(_Float16));
  float* dw1w = fw;             // 64*7
  float* dw2w = fw + 448;       // 64*7
  float* s1 = fw + 896;  float* b1 = fw + 960;
  float* s2 = fw + 1024; float* b2 = fw + 1088;
  float* pwrL = fw + 1152;      // [2][64]
  float* pweL = fw + 1280;      // [64][2]
  float* seL  = fw + 1408;      // 64
  float* beL  = fw + 1472;      // 64
  float* bnrL = fw + 1536;      // s0,s1,b0,b1
  float* csum = fw + 1540;      // 64 (ECA partial channel sums)

  int n = blockIdx.y;
  int y0 = blockIdx.x * RROWS;
  int t = threadIdx.x;
  int bsel = idx[n];
  int k = 3 + 2 * bsel;
  const float* dw1 = P.p[bsel * 8 + 0];
  const float* bn1 = P.p[bsel * 8 + 1];
  const float* dw2 = P.p[bsel * 8 + 2];
  const float* bn2 = P.p[bsel * 8 + 3];
  const float* pwr = P.p[bsel * 8 + 4];
  const float* bnr = P.p[bsel * 8 + 5];
  const float* pwe = P.p[bsel * 8 + 6];
  const float* bne = P.p[bsel * 8 + 7];

  if (t < CG) {
    int c = t;
    {
      float g = bn1[c], bb = bn1[64 + c], mm = bn1[128 + c], vv = bn1[192 + c];
      float sc = g * rsqrtf(vv + BN_EPS); s1[c] = sc; b1[c] = bb - mm * sc;
    }
    {
      float g = bn2[c], bb = bn2[64 + c], mm = bn2[128 + c], vv = bn2[192 + c];
      float sc = g * rsqrtf(vv + BN_EPS); s2[c] = sc; b2[c] = bb - mm * sc;
    }
    {
      float g = bne[c], bb = bne[64 + c], mm = bne[128 + c], vv = bne[192 + c];
      float sc = g * rsqrtf(vv + BN_EPS); seL[c] = sc; beL[c] = bb - mm * sc;
    }
    for (int j = 0; j < k; ++j) { dw1w[c * 7 + j] = dw1[c * k + j]; dw2w[c * 7 + j] = dw2[c * k + j]; }
    pwrL[c] = pwr[c]; pwrL[64 + c] = pwr[64 + c];
    pweL[c] = pwe[c]; pweL[64 + c] = pwe[64 + c];
    csum[c] = 0.f;
    if (c < 2) {
      float g = bnr[c], bb = bnr[2 + c], mm = bnr[4 + c], vv = bnr[6 + c];
      float sc = g * rsqrtf(vv + BN_EPS); bnrL[c] = sc; bnrL[2 + c] = bb - mm * sc;
    }
  }
  __syncthreads();

  // Phase 1: DW 1xk (dilation 2) + BN + ReLU -> h1 in LDS (f16), incl. halo rows
  const int TOT = LDSROWS * CG * WW;  // 102400
  for (int e = t; e < TOT; e += 256) {
    int rl = e / (CG * WW);
    int rem = e - rl * (CG * WW);
    int c = rem / WW;
    int px = rem - c * WW;
    int ys = y0 - HALO + rl;
    float val = 0.f;  // zero pad rows (matches conv zero padding of h1)
    if (ys >= 0 && ys < HH) {
      const float* row = x + (((size_t)(n * CG + c)) * HH + ys) * WW;
      float s = 0.f;
      for (int j = 0; j < k; ++j) {
        int xs = px + 2 * j - (k - 1);
        if (xs >= 0 && xs < WW) s += dw1w[c * 7 + j] * row[xs];
      }
      val = fmaxf(s * s1[c] + b1[c], 0.f);
    }
    h1[(rl * WW + px) * STR + c] = (_Float16)val;
  }
  __syncthreads();

  // Phase 2: per wave, 16-pixel groups: DW kx1 + BN + ReLU -> h2 on the fly,
  // rank-2 channel reduce via V_WMMA_F32_16X16X4_F32 (K=64 in 16 steps),
  // then rank-2 expand + BN + ReLU, store + ECA channel sums.
  int wave = t >> 5;
  int lane = t & 31;
  int m = lane & 15;
  int chalf = (lane >> 4) << 1;  // lanes 0-15: K=0,1 ; lanes 16-31: K=2,3 (ISA layout)
  for (int g = wave; g < (RROWS * WW) / 16; g += 8) {   // 40 groups, 5 per wave
    int p = g * 16 + m;
    int py = p / WW;
    int px = p - py * WW;
    v8f acc = {};
    for (int kb = 0; kb < CG; kb += 4) {
      int c0 = kb + chalf, c1 = c0 + 1;
      float sa = 0.f, sb = 0.f;
      for (int j = 0; j < k; ++j) {
        int rl = py + HALO + 2 * j - (k - 1);
        v2h hv = *(const v2h*)&h1[(rl * WW + px) * STR + c0];  // both channels, 1 b32 load
        sa += dw2w[c0 * 7 + j] * (float)hv.x;
        sb += dw2w[c1 * 7 + j] * (float)hv.y;
      }
      float ha = fmaxf(sa * s2[c0] + b2[c0], 0.f);
      float hb = fmaxf(sb * s2[c1] + b2[c1], 0.f);
      v2f bfrag; bfrag.x = ha; bfrag.y = hb;            // B[4x16]: h2 channels x pixels
      v2f afrag;                                         // A[16x4]: pwr rows 0,1; rest 0
      afrag.x = (m < 2) ? pwrL[m * 64 + c0] : 0.f;
      afrag.y = (m < 2) ? pwrL[m * 64 + c1] : 0.f;
      acc = __builtin_amdgcn_wmma_f32_16x16x4_f32(false, afrag, false, bfrag,
                                                  (short)0, acc, false, false);
    }
    // D rows 0,1 (= r0,r1 per pixel) live in acc[0],acc[1] of lanes 0-15
    float r0 = __shfl(acc[0], m, 32);
    float r1 = __shfl(acc[1], m, 32);
    r0 = fmaxf(r0 * bnrL[0] + bnrL[2], 0.f);
    r1 = fmaxf(r1 * bnrL[1] + bnrL[3], 0.f);
    int cbase = (lane >> 4) * 32;  // lanes 0-15: c 0..31, lanes 16-31: c 32..63
    size_t obase = ((size_t)(n * CG)) * HH * WW + (size_t)(y0 + py) * WW + px;
    for (int ci = 0; ci < 32; ++ci) {
      int c = cbase + ci;
      float o = pweL[c * 2] * r0 + pweL[c * 2 + 1] * r1;
      o = fmaxf(o * seL[c] + beL[c], 0.f);
      out[obase + (size_t)c * (HH * WW)] = o;
      float v = o;                      // butterfly sum over the 16 pixels (same c)
      v += __shfl_xor(v, 1, 32);
      v += __shfl_xor(v, 2, 32);
      v += __shfl_xor(v, 4, 32);
      v += __shfl_xor(v, 8, 32);
      if (m == 0) atomicAdd(&csum[c], v);
    }
  }
  __syncthreads();
  if (t < CG) atomicAdd(&chansum_g[n * CG + t], csum[t]);
}

// ---------------------------------------------------------------- ECA gating
__global__ void eca_kernel(const float* __restrict__ chansum,
                           const float* __restrict__ eca_w,
                           float* __restrict__ gate) {
  int n = blockIdx.x, c = threadIdx.x;
  const float invHW = 1.f / 6400.f;
  float ym1 = (c > 0)  ? chansum[n * CG + c - 1] * invHW : 0.f;
  float yc  = chansum[n * CG + c] * invHW;
  float yp1 = (c < 63) ? chansum[n * CG + c + 1] * invHW : 0.f;
  float a = eca_w[0] * ym1 + eca_w[1] * yc + eca_w[2] * yp1;
  gate[n * CG + c] = 1.f / (1.f + __expf(-a));
}

// ------------------------------------------------------- in-place gate scale
__global__ void scale_kernel(float* __restrict__ out, const float* __restrict__ gate) {
  const size_t tot4 = (size_t)NGRP * CG * HH * WW / 4;
  float4* p = (float4*)out;
  for (size_t i = (size_t)blockIdx.x * blockDim.x + threadIdx.x; i < tot4;
       i += (size_t)gridDim.x * blockDim.x) {
    int nc = (int)(i / (HH * WW / 4));
    float g = gate[nc];
    float4 v = p[i];
    v.x *= g; v.y *= g; v.z *= g; v.w *= g;
    p[i] = v;
  }
}

extern "C" void kernel_launch(void* const* d_in, const int* in_sizes, int n_in,
                              void* d_out, int out_size, void* d_ws, size_t ws_size,
                              hipStream_t stream) {
  (void)in_sizes; (void)n_in; (void)out_size; (void)ws_size;
  const float* x     = (const float*)d_in[0];
  const float* sel_w = (const float*)d_in[1];
  const float* sel_b = (const float*)d_in[2];
  Params24 P;
  for (int i = 0; i < 24; ++i) P.p[i] = (const float*)d_in[3 + i];
  const float* eca_w = (const float*)d_in[27];
  float* out = (float*)d_out;

  char* ws = (char*)d_ws;
  float* chansum = (float*)ws;                 // 4096 floats
  float* gate    = (float*)(ws + 4096 * 4);    // 4096 floats
  int*   idx     = (int*)(ws + 8192 * 4);      // 64 ints

  zero_kernel<<<16, 256, 0, stream>>>(chansum, NGRP * CG);
  select_kernel<<<NGRP, 256, 0, stream>>>(x, sel_w, sel_b, idx);
  size_t shmem = (size_t)LDSROWS * WW * STR * sizeof(_Float16) + 1604 * sizeof(float);
  branch_kernel<<<dim3(HH / RROWS, NGRP), 256, shmem, stream>>>(x, P, idx, out, chansum);
  eca_kernel<<<NGRP, CG, 0, stream>>>(chansum, eca_w, gate);
  scale_kernel<<<2048, 256, 0, stream>>>(out, gate);
}